// BilateralFilter_5042291605780
// MI455X (gfx1250) — compile-verified
//
#include <hip/hip_runtime.h>

// Bilateral 5x5, sigma_color=0.1, sigma_spatial=1, zero padding (skimage constant mode).
// CDNA5: async global->LDS scatter staging (ASYNCcnt) into 16B/pixel AoS records
// [r,g,b,|p|^2]; center tap analytic (w=1); 24 taps, each = 1 ds_load_b128 with
// immediate offset + 9 scalar VALU + 1 v_exp_f32.

#define IMG_H   512
#define IMG_W   512
#define NCH     3
#define RAD     2
#define TILE    32
#define TDIM    (TILE + 2 * RAD)          // 36
#define NPIX    (TDIM * TDIM)             // 1296 tile pixels
#define NXFER   (NPIX * NCH)              // 3888 dword transfers
#define NTHR    256                        // 8 wave32 per block
#define LOG2E   1.4426950408889634f
#define KC      (-50.0f * LOG2E)          // exp2 scale for color term
#define M2KC    (100.0f * LOG2E)          // -2*KC
#define SCONST(s) (-0.5f * LOG2E * (float)(s))   // spatial log2-weight, s = dy^2+dx^2

#if __has_builtin(__builtin_amdgcn_exp2f)
#define FAST_EXP2(x) __builtin_amdgcn_exp2f(x)
#else
#define FAST_EXP2(x) exp2f(x)
#endif

struct __align__(16) F4 { float r, g, b, q; };   // q = r^2+g^2+b^2
struct F3 { float r, g, b; };

__global__ __launch_bounds__(NTHR)
void bilateral5x5_kernel(const float* __restrict__ xin, float* __restrict__ out)
{
    __shared__ F4 tile4[NPIX];            // 20736 bytes of 320KB/WGP

    const int t  = threadIdx.x;
    const int x0 = blockIdx.x * TILE;
    const int y0 = blockIdx.y * TILE;
    const int b  = blockIdx.z;

    // LDS byte offset: generic LDS addresses truncate to addr[31:0] (ISA 10.2).
    const unsigned lds_base = (unsigned)(unsigned long long)(const void*)tile4;
    const unsigned long long gbase = (unsigned long long)xin;
    float* ldsf = (float*)tile4;

    // ---- Stage haloed tile via CDNA5 async global->LDS scatter ----
    // Repacks 12B/pixel global RGB into 16B/pixel LDS records (q slot filled later).
    auto stage = [&](int d) {
        int pix = d / NCH;
        int ch  = d - pix * NCH;
        int r   = pix / TDIM;
        int px  = pix - r * TDIM;
        int gy  = y0 - RAD + r;
        int gx  = x0 - RAD + px;
        unsigned lds_addr = lds_base + (unsigned)(pix * 16 + ch * 4);
        if (gy >= 0 && gy < IMG_H && gx >= 0 && gx < IMG_W) {
            int goff = ((((b * IMG_H) + gy) * IMG_W + gx) * NCH + ch) * 4;
            asm volatile("global_load_async_to_lds_b32 %0, %1, %2"
                         :
                         : "v"(lds_addr), "v"(goff), "s"(gbase)
                         : "memory");
        } else {
            ldsf[pix * 4 + ch] = 0.0f;    // constant (zero) padding
        }
    };
    #pragma unroll
    for (int k = 0; k < NXFER / NTHR; ++k)      // 15 full iterations: d < 3840+256 bound-safe
        stage(t + k * NTHR);
    {                                            // tail: only t < NXFER - 15*NTHR participate
        int d = t + (NXFER / NTHR) * NTHR;
        if (d < NXFER) stage(d);
    }
    asm volatile("s_wait_asynccnt 0x0" ::: "memory");
    __syncthreads();

    // ---- Fill the q = |p|^2 plane (zeros stay zero) ----
    auto fillq = [&](int pix) {
        float pr = ldsf[pix * 4 + 0];
        float pg = ldsf[pix * 4 + 1];
        float pb = ldsf[pix * 4 + 2];
        ldsf[pix * 4 + 3] = fmaf(pr, pr, fmaf(pg, pg, pb * pb));
    };
    #pragma unroll
    for (int k = 0; k < NPIX / NTHR; ++k)        // 5 full iterations
        fillq(t + k * NTHR);
    {
        int pix = t + (NPIX / NTHR) * NTHR;
        if (pix < NPIX) fillq(pix);
    }
    __syncthreads();

    // 24 non-center taps, offsets relative to window top-left (all non-negative ->
    // fold into unsigned DS immediate offsets). j = (dy+2)*36 + (dx+2), center j=74.
    constexpr int JOFF[24] = {
          0,   1,   2,   3,   4,
         36,  37,  38,  39,  40,
         72,  73,       75,  76,
        108, 109, 110, 111, 112,
        144, 145, 146, 147, 148 };
    constexpr int SS[24] = {
        8, 5, 4, 5, 8,
        5, 2, 1, 2, 5,
        4, 1,    1, 4,
        5, 2, 1, 2, 5,
        8, 5, 4, 5, 8 };

    F3* op = (F3*)out;

    // ---- Compute: 4 output pixels per thread, fully unrolled 24-tap window ----
    #pragma unroll
    for (int k = 0; k < (TILE * TILE) / NTHR; ++k) {
        int p  = t + k * NTHR;
        int py = p >> 5;
        int px = p & (TILE - 1);
        int cy = py + RAD;
        int cx = px + RAD;
        const F4* wb = tile4 + (cy * TDIM + cx - 74);   // window top-left

        F4 cc = wb[74];
        // As[s] = KC*|c|^2 + spatial_const(s),  s = dy^2+dx^2 in {1,2,4,5,8}
        float A = cc.q * KC;
        float As[9];
        As[1] = A + SCONST(1);
        As[2] = A + SCONST(2);
        As[4] = A + SCONST(4);
        As[5] = A + SCONST(5);
        As[8] = A + SCONST(8);
        As[0] = As[3] = As[6] = As[7] = A;   // unused

        // Seed with the center tap: weight exactly 1.
        float sr = cc.r, sg = cc.g, sb = cc.b, sw = 1.0f;

        #pragma unroll
        for (int i = 0; i < 24; ++i) {
            F4 nv = wb[JOFF[i]];                       // ds_load_b128, imm offset
            // |n-c|^2 = n.q + c.q - 2*dot(n,c); exponent in log2 domain:
            // arg = KC*n.q + (KC*c.q + sconst) + M2KC*dot
            float dot = fmaf(nv.r, cc.r, fmaf(nv.g, cc.g, nv.b * cc.b));
            float arg = fmaf(nv.q, KC, As[SS[i]]);
            arg = fmaf(dot, M2KC, arg);
            float w = FAST_EXP2(arg);                  // v_exp_f32
            sr = fmaf(w, nv.r, sr);
            sg = fmaf(w, nv.g, sg);
            sb = fmaf(w, nv.b, sb);
            sw += w;
        }

        float inv = __builtin_amdgcn_rcpf(sw);   // den >= 1 (center weight is 1)
        int gp = ((b * IMG_H) + (y0 + py)) * IMG_W + (x0 + px);
        F3 o;
        o.r = sr * inv;
        o.g = sg * inv;
        o.b = sb * inv;
        op[gp] = o;
    }
}

extern "C" void kernel_launch(void* const* d_in, const int* in_sizes, int n_in,
                              void* d_out, int out_size, void* d_ws, size_t ws_size,
                              hipStream_t stream)
{
    (void)n_in; (void)out_size; (void)d_ws; (void)ws_size;
    const float* x = (const float*)d_in[0];
    float* outp = (float*)d_out;
    int B = in_sizes[0] / (IMG_H * IMG_W * NCH);   // 16 for the reference shapes
    dim3 grid(IMG_W / TILE, IMG_H / TILE, B);
    bilateral5x5_kernel<<<grid, NTHR, 0, stream>>>(x, outp);
}